// Lstm_65506841199249
// MI455X (gfx1250) — compile-verified
//
#include <hip/hip_runtime.h>

typedef __attribute__((ext_vector_type(2))) float v2f;
typedef __attribute__((ext_vector_type(4))) float v4f;
typedef __attribute__((ext_vector_type(8))) float v8f;

#define Hh 16
#define T1 96
#define T2 48
#define BN 256
#define LL 24

static __device__ __forceinline__ float sigmf(float x) {
    return 1.0f / (1.0f + __expf(-x));
}
static __device__ __forceinline__ float tanh_fast(float x) {
    float ax = fabsf(x);
    float e = __expf(-2.0f * ax);
    float t = (1.0f - e) / (1.0f + e);
    return copysignf(t, x);
}

// -------------------------------------------------------------------------
// Generic f32 WMMA GEMM:  out[m,n] = sum_k A[m,k] * W[n,k] + bias0[n]+bias1[n]
// A: (M x K) row-major, row stride lda. W: (N x K) row-major, stride ldw.
// grid = (M/16, ceil(N/16)), block = 32 (one wave).
// tmode == 0: out is (M x ldo) row-major (per-element stores, n-guarded).
// tmode == 1: xg-transposed store: out[((m/BN)*64 + n)*BN + (m%BN)],
//             i.e. layout (T, 64, BN); per-lane contiguous -> 2x b128 stores.
// Branch-free K-loop: invalid n lanes clamp to row N-1 (discarded at store).
// -------------------------------------------------------------------------
__global__ void wmma_gemm_kernel(const float* __restrict__ A, int lda,
                                 const float* __restrict__ W, int ldw,
                                 const float* __restrict__ bias0,
                                 const float* __restrict__ bias1,
                                 float* __restrict__ out, int ldo,
                                 int M, int N, int K, int relu, int tmode) {
    const int lane = threadIdx.x;
    const int hl   = lane >> 4;       // 0/1
    const int l15  = lane & 15;
    const int mt   = blockIdx.x;
    const int nt   = blockIdx.y;

    const int arow = mt * 16 + l15;   // A layout: lane = row
    const int ncol = nt * 16 + l15;   // B/C/D layout: lane = col
    const bool nvalid = (ncol < N);
    const int ncl = nvalid ? ncol : (N - 1);   // clamp: keeps loop branch-free

    const float* __restrict__ Arow = A + (size_t)arow * lda;
    const float* __restrict__ Wrow = W + (size_t)ncl * ldw;

    v8f c = {};
#pragma unroll 4
    for (int k0 = 0; k0 < K; k0 += 4) {
        const int ka = k0 + 2 * hl;                 // K pair for this lane half
        v2f a = *(const v2f*)(Arow + ka);
        v2f b = *(const v2f*)(Wrow + ka);
        c = __builtin_amdgcn_wmma_f32_16x16x4_f32(false, a, false, b,
                                                  (short)0, c, false, false);
    }

    float bs = 0.0f;
    if (nvalid) {
        if (bias0) bs += bias0[ncol];
        if (bias1) bs += bias1[ncol];
    }

    if (tmode) {
        // (T, 64, BN) store: rows mt*16+hl*8 .. +7 are contiguous in the b index.
        const int row0 = mt * 16 + hl * 8;
        const int t  = row0 / BN;
        const int b0 = row0 % BN;
        float* op = out + ((size_t)t * 64 + ncol) * BN + b0;
        v4f s0, s1;
        s0.x = c[0] + bs; s0.y = c[1] + bs; s0.z = c[2] + bs; s0.w = c[3] + bs;
        s1.x = c[4] + bs; s1.y = c[5] + bs; s1.z = c[6] + bs; s1.w = c[7] + bs;
        *(v4f*)op       = s0;
        *(v4f*)(op + 4) = s1;
    } else {
#pragma unroll
        for (int r = 0; r < 8; ++r) {
            float v = c[r] + bs;
            if (relu) v = fmaxf(v, 0.0f);
            const int row = mt * 16 + hl * 8 + r;
            if (nvalid) out[(size_t)row * ldo + ncol] = v;
        }
    }
}

// -------------------------------------------------------------------------
// Layer-0 (input dim 1) gate precompute into transposed xg layout (d,T,64,BN):
//   xg[d][(t*64+g)*BN + b] = x[b,t]*Wih0[d,g] + bih[d,g] + bhh[d,g]
// -------------------------------------------------------------------------
__global__ void l0_gates_kernel(const float* __restrict__ x,      // (B,1,T)
                                const float* __restrict__ Wih0,   // (2,64,1)
                                const float* __restrict__ bih,    // (L,2,64) layer 0
                                const float* __restrict__ bhh,
                                float* __restrict__ xg, int T) {
    int idx = blockIdx.x * blockDim.x + threadIdx.x;
    int total = 2 * T * 64 * BN;
    if (idx >= total) return;
    int b  = idx & (BN - 1);
    int r1 = idx >> 8;
    int g  = r1 & 63;
    int r2 = r1 >> 6;
    int t  = r2 % T;
    int d  = r2 / T;
    float xv = x[(size_t)b * T + t];
    xg[idx] = xv * Wih0[d * 64 + g] + bih[d * 64 + g] + bhh[d * 64 + g];
}

// -------------------------------------------------------------------------
// LSTM sequential scan (one layer; grid = (BN/16, 2 dirs), block = 1 wave).
// xg layout (2, T, 64, BN): per-lane gate-tile loads are 2x b128, and the
// next timestep's tile is prefetched into registers (software pipelining)
// so its latency hides behind the 16 WMMAs + gate VALU of the current step.
//   Whh : (2, 64, 16)    xout : (T, BN, 32)
// -------------------------------------------------------------------------
__global__ void lstm_scan_kernel(const float* __restrict__ xg,
                                 const float* __restrict__ Whh,
                                 float* __restrict__ xout,
                                 int T) {
    __shared__ float hbuf[16 * 20];   // 16x16 tile, stride 20 (16B-aligned b128)
    const int lane = threadIdx.x;
    const int hl   = lane >> 4;
    const int l15  = lane & 15;
    const int mt   = blockIdx.x;      // batch tile
    const int dir  = blockIdx.y;      // 0 = fwd, 1 = bwd

    const float* __restrict__ xgd = xg + (size_t)dir * T * 64 * BN;
    const float* __restrict__ Wd  = Whh + dir * 64 * 16;

    // Preload Whh^T as 4 gate-tiles x 4 K-chunks in B layout (loop invariant).
    v2f wb[4][4];
#pragma unroll
    for (int gt = 0; gt < 4; ++gt)
#pragma unroll
        for (int kc = 0; kc < 4; ++kc) {
            const int n = gt * 16 + l15;          // gate index (lane = col)
            const int k = kc * 4 + 2 * hl;        // hidden K pair
            wb[gt][kc] = *(const v2f*)(Wd + n * 16 + k);
        }

    v8f cst = {};                                  // cell state (D layout)
    v2f hA[4];                                     // h in A layout per K-chunk
#pragma unroll
    for (int kc = 0; kc < 4; ++kc) { hA[kc].x = 0.0f; hA[kc].y = 0.0f; }

    // xg tile lane address: gate col = gt*16 + l15, 8 contiguous batch rows.
    const int bbase = mt * 16 + hl * 8;

    // Prefetch first timestep.
    v8f nxt[4];
    {
        const int t0 = (dir == 0) ? 0 : (T - 1);
        const float* xgt = xgd + (size_t)t0 * 64 * BN + bbase;
#pragma unroll
        for (int gt = 0; gt < 4; ++gt) {
            const float* p = xgt + (size_t)(gt * 16 + l15) * BN;
            v4f lo = *(const v4f*)p;
            v4f hi = *(const v4f*)(p + 4);
            nxt[gt][0] = lo.x; nxt[gt][1] = lo.y; nxt[gt][2] = lo.z; nxt[gt][3] = lo.w;
            nxt[gt][4] = hi.x; nxt[gt][5] = hi.y; nxt[gt][6] = hi.z; nxt[gt][7] = hi.w;
        }
    }

    for (int s = 0; s < T; ++s) {
        const int t = (dir == 0) ? s : (T - 1 - s);

        // Consume prefetched xg tile.
        v8f acc[4];
#pragma unroll
        for (int gt = 0; gt < 4; ++gt) acc[gt] = nxt[gt];

        // Prefetch next timestep (independent of this step's recurrence).
        if (s + 1 < T) {
            const int tn = (dir == 0) ? (s + 1) : (T - 2 - s);
            const float* xgt = xgd + (size_t)tn * 64 * BN + bbase;
#pragma unroll
            for (int gt = 0; gt < 4; ++gt) {
                const float* p = xgt + (size_t)(gt * 16 + l15) * BN;
                v4f lo = *(const v4f*)p;
                v4f hi = *(const v4f*)(p + 4);
                nxt[gt][0] = lo.x; nxt[gt][1] = lo.y; nxt[gt][2] = lo.z; nxt[gt][3] = lo.w;
                nxt[gt][4] = hi.x; nxt[gt][5] = hi.y; nxt[gt][6] = hi.z; nxt[gt][7] = hi.w;
            }
        }

        // acc += h (16x16) x Whh^T (16x64): 4 N-tiles x 4 K-chunks of f32 WMMA.
#pragma unroll
        for (int gt = 0; gt < 4; ++gt)
#pragma unroll
            for (int kc = 0; kc < 4; ++kc)
                acc[gt] = __builtin_amdgcn_wmma_f32_16x16x4_f32(
                    false, hA[kc], false, wb[gt][kc], (short)0, acc[gt],
                    false, false);

        // Gates i,f,g,o: c = sig(f)*c + sig(i)*tanh(g); h = sig(o)*tanh(c).
        v8f hD;
#pragma unroll
        for (int r = 0; r < 8; ++r) {
            const float iv = sigmf(acc[0][r]);
            const float fv = sigmf(acc[1][r]);
            const float gv = tanh_fast(acc[2][r]);
            const float ov = sigmf(acc[3][r]);
            const float cv = fv * cst[r] + iv * gv;
            cst[r] = cv;
            hD[r]  = ov * tanh_fast(cv);
        }

        // D -> LDS tile: hbuf[m*20 + n] = h[m][n]
        __syncthreads();
#pragma unroll
        for (int r = 0; r < 8; ++r) {
            const int row = hl * 8 + r;
            hbuf[row * 20 + l15] = hD[r];
        }
        __syncthreads();

        // h back out of LDS in A layout for the next step's WMMAs.
#pragma unroll
        for (int kc = 0; kc < 4; ++kc) {
            const int k = kc * 4 + 2 * hl;
            hA[kc] = *(const v2f*)(&hbuf[l15 * 20 + k]);
        }

        // Coalesced h output: 2 lanes per batch row, 2x b128 per lane.
        {
            const int srow = lane >> 1;            // 0..15
            const int scol = (lane & 1) * 8;       // 0 or 8
            v4f s0 = *(const v4f*)(&hbuf[srow * 20 + scol]);
            v4f s1 = *(const v4f*)(&hbuf[srow * 20 + scol + 4]);
            float* op = xout + ((size_t)t * BN + mt * 16 + srow) * 32
                             + dir * 16 + scol;
            *(v4f*)op       = s0;
            *(v4f*)(op + 4) = s1;
        }
    }
}

// -------------------------------------------------------------------------
// Residual + flatten transpose: zb[b, t*32+j] = h2[t,b,j] + x1[t,b,j]
// -------------------------------------------------------------------------
__global__ void residual_flatten_kernel(const float* __restrict__ h2,
                                        const float* __restrict__ x1,
                                        float* __restrict__ zb,
                                        int T) {
    int idx = blockIdx.x * blockDim.x + threadIdx.x;
    int total = T * BN * 32;
    if (idx >= total) return;
    int j  = idx & 31;
    int r1 = idx >> 5;
    int b  = r1 & (BN - 1);
    int t  = r1 >> 8;
    zb[(size_t)b * (T * 32) + t * 32 + j] = h2[idx] + x1[idx];
}

// -------------------------------------------------------------------------
extern "C" void kernel_launch(void* const* d_in, const int* in_sizes, int n_in,
                              void* d_out, int out_size, void* d_ws, size_t ws_size,
                              hipStream_t stream) {
    const float* x      = (const float*)d_in[0];   // (256,1,96)
    const float* Wih0_1 = (const float*)d_in[1];   // (2,64,1)
    const float* Wih_1  = (const float*)d_in[2];   // (23,2,64,32)
    const float* Whh_1  = (const float*)d_in[3];   // (24,2,64,16)
    const float* bih_1  = (const float*)d_in[4];   // (24,2,64)
    const float* bhh_1  = (const float*)d_in[5];   // (24,2,64)
    const float* Wih_2  = (const float*)d_in[6];   // (24,2,64,32)
    const float* Whh_2  = (const float*)d_in[7];   // (24,2,64,16)
    const float* bih_2  = (const float*)d_in[8];   // (24,2,64)
    const float* bhh_2  = (const float*)d_in[9];   // (24,2,64)
    const float* fc1_w  = (const float*)d_in[10];  // (1000,1536)
    const float* fc1_b  = (const float*)d_in[11];  // (1000)
    const float* fc2_w  = (const float*)d_in[12];  // (48,1000)
    const float* fc2_b  = (const float*)d_in[13];  // (48)
    float* out = (float*)d_out;                    // (256,1,48)

    float* ws    = (float*)d_ws;
    float* xg    = ws;                                   // 2*96*64*256
    float* bufX  = xg    + (size_t)2 * T1 * 64 * BN;     // 96*256*32
    float* bufX1 = bufX  + (size_t)T1 * BN * 32;         // 48*256*32
    float* zb    = bufX1 + (size_t)T2 * BN * 32;         // 256*1536
    float* f1    = zb    + (size_t)BN * (T2 * 32);       // 256*1000

    const dim3 blk32(32, 1, 1);
    const dim3 scan_grid(BN / 16, 2, 1);

    // -------- Stack 1, layer 0 (input dim = 1) --------
    {
        int total = 2 * T1 * 64 * BN;
        l0_gates_kernel<<<(total + 255) / 256, 256, 0, stream>>>(
            x, Wih0_1, bih_1, bhh_1, xg, T1);
        lstm_scan_kernel<<<scan_grid, blk32, 0, stream>>>(xg, Whh_1, bufX, T1);
    }
    // -------- Stack 1, layers 1..23 (input dim = 32) --------
    for (int l = 1; l < LL; ++l) {
        for (int d = 0; d < 2; ++d) {
            wmma_gemm_kernel<<<dim3(T1 * BN / 16, 4), blk32, 0, stream>>>(
                bufX, 32,
                Wih_1 + ((size_t)(l - 1) * 2 + d) * 64 * 32, 32,
                bih_1 + ((size_t)l * 2 + d) * 64,
                bhh_1 + ((size_t)l * 2 + d) * 64,
                xg + (size_t)d * T1 * 64 * BN, 0,
                T1 * BN, 64, 32, 0, 1);
        }
        lstm_scan_kernel<<<scan_grid, blk32, 0, stream>>>(
            xg, Whh_1 + (size_t)l * 2 * 64 * 16, bufX, T1);
    }

    // -------- x1 = last 48 timesteps (saved for residual) --------
    float* p2 = bufX + (size_t)T2 * BN * 32;   // tail of stack-1 output, (48,256,32)
    hipMemcpyAsync(bufX1, p2, (size_t)T2 * BN * 32 * sizeof(float),
                   hipMemcpyDeviceToDevice, stream);

    // -------- Stack 2, layers 0..23 (input dim = 32, T = 48, in-place) --------
    for (int l = 0; l < LL; ++l) {
        for (int d = 0; d < 2; ++d) {
            wmma_gemm_kernel<<<dim3(T2 * BN / 16, 4), blk32, 0, stream>>>(
                p2, 32,
                Wih_2 + ((size_t)l * 2 + d) * 64 * 32, 32,
                bih_2 + ((size_t)l * 2 + d) * 64,
                bhh_2 + ((size_t)l * 2 + d) * 64,
                xg + (size_t)d * T2 * 64 * BN, 0,
                T2 * BN, 64, 32, 0, 1);
        }
        lstm_scan_kernel<<<scan_grid, blk32, 0, stream>>>(
            xg, Whh_2 + (size_t)l * 2 * 64 * 16, p2, T2);
    }

    // -------- Residual + flatten to (B, 1536) --------
    {
        int total = T2 * BN * 32;
        residual_flatten_kernel<<<(total + 255) / 256, 256, 0, stream>>>(
            p2, bufX1, zb, T2);
    }

    // -------- FC1 (1536 -> 1000) + ReLU --------
    wmma_gemm_kernel<<<dim3(BN / 16, (1000 + 15) / 16), blk32, 0, stream>>>(
        zb, T2 * 32, fc1_w, T2 * 32, fc1_b, nullptr,
        f1, 1000, BN, 1000, T2 * 32, 1, 0);

    // -------- FC2 (1000 -> 48), straight to d_out (B,1,48) --------
    wmma_gemm_kernel<<<dim3(BN / 16, 3), blk32, 0, stream>>>(
        f1, 1000, fc2_w, 1000, fc2_b, nullptr,
        out, T2, BN, T2, 1000, 0, 0);
}